// CandiRec_19078244728940
// MI455X (gfx1250) — compile-verified
//
#include <hip/hip_runtime.h>
#include <hip/hip_bf16.h>
#include <math.h>

// ---------------------------------------------------------------------------
// CandiRec on MI455X (gfx1250, wave32, WMMA + async global->LDS staging).
// Pipeline:
//   init  : zero histograms / counters / loss accumulator
//   mlp1  : h = q @ W1 + b1            (WMMA f32 16x16x4)
//   mlp2  : query = h @ W2 + b2        (WMMA f32 16x16x4)
//   l2n   : qn = l2norm(query)
//   sims<0>: sims (WMMA, async-LDS staged emb) + per-row 64-bin histograms
//   thr   : per-row top-50 threshold from histogram
//   sims<1>: recompute sims, collect >= threshold candidates
//   final : per-row exact sorted top-50, fixed_idx, CE + contrastive loss
// ---------------------------------------------------------------------------

typedef __attribute__((ext_vector_type(2))) float v2f;
typedef __attribute__((ext_vector_type(8))) float v8f;

#define BQ   64      // batch (query rows)
#define DIM  128     // embedding dim
#define LD   512     // hidden dim
#define LQ   4096    // query feature dim
#define TOPK 20
#define MAXK 50
#define NBIN 64
#define CAP  4096    // candidate capacity per row
#define CHUNK 1024   // items per workgroup in sims kernels
#define TI   32      // items per double-buffer stage
#define EPAD (DIM + 4)  // padded LDS row (floats): bank-conflict-free ds_load_b64

__device__ __forceinline__ v8f wmma_f32(v2f a, v2f b, v8f c) {
  // D = A(16x4,f32) * B(4x16,f32) + C(16x16,f32)
  return __builtin_amdgcn_wmma_f32_16x16x4_f32(
      /*neg_a=*/false, a, /*neg_b=*/false, b,
      /*c_mod=*/(short)0, c, /*reuse_a=*/false, /*reuse_b=*/false);
}

// Async-copy 64B (4 x b128) of one embedding row chunk into LDS.
// IOFFSET is added to BOTH the global and LDS address (CDNA5 ISA).
__device__ __forceinline__ void async_copy64(const void* g, void* ldsDst) {
  unsigned lds = (unsigned)(uintptr_t)ldsDst;  // low 32 bits = LDS offset
  asm volatile(
      "global_load_async_to_lds_b128 %0, %1, off\n\t"
      "global_load_async_to_lds_b128 %0, %1, off offset:16\n\t"
      "global_load_async_to_lds_b128 %0, %1, off offset:32\n\t"
      "global_load_async_to_lds_b128 %0, %1, off offset:48"
      :: "v"(lds), "v"(g) : "memory");
}

// Fixed stand-in for jax.random.permutation(key(0), 20).
__constant__ int PERM20[TOPK] = {3, 17, 8, 0, 12, 19, 5, 14, 1, 10,
                                 7, 16, 2, 11, 18, 6, 13, 4, 15, 9};

// ---------------------------------------------------------------- init ----
__global__ void k_init(unsigned int* hist, int* cnt, float* lossOut) {
  int t = blockIdx.x * blockDim.x + threadIdx.x;
  if (t < BQ * NBIN) hist[t] = 0u;
  if (t < BQ) cnt[t] = 0;
  if (t == 0) *lossOut = 0.0f;
}

// ------------------------------------------------------- MLP layer 1 ------
// h[64x512] = q[64x4096] @ W1[4096x512] + b1 ; one wave per 16x16 tile.
__global__ __launch_bounds__(32) void k_mlp1(const float* __restrict__ q,
                                             const float* __restrict__ W1,
                                             const float* __restrict__ b1,
                                             float* __restrict__ h) {
  int lane = threadIdx.x & 31, l15 = lane & 15, hi = lane >> 4;
  int mT = blockIdx.y, nT = blockIdx.x;
  int row = mT * 16 + l15, col = nT * 16 + l15;
  const float* ap = q + (size_t)row * LQ + hi * 2;
  v8f c = {};
  for (int k = 0; k < LQ; k += 4) {
    v2f a = *(const v2f*)(ap + k);
    int kk = k + hi * 2;
    v2f b;
    b.x = W1[(size_t)kk * LD + col];
    b.y = W1[(size_t)(kk + 1) * LD + col];
    c = wmma_f32(a, b, c);
  }
#pragma unroll
  for (int r = 0; r < 8; ++r) {
    int rr = mT * 16 + r + hi * 8;
    h[(size_t)rr * LD + col] = c[r] + b1[col];
  }
}

// ------------------------------------------------------- MLP layer 2 ------
// query[64x128] = h[64x512] @ W2[512x128] + b2
__global__ __launch_bounds__(32) void k_mlp2(const float* __restrict__ h,
                                             const float* __restrict__ W2,
                                             const float* __restrict__ b2,
                                             float* __restrict__ qry) {
  int lane = threadIdx.x & 31, l15 = lane & 15, hi = lane >> 4;
  int mT = blockIdx.y, nT = blockIdx.x;
  int row = mT * 16 + l15, col = nT * 16 + l15;
  const float* ap = h + (size_t)row * LD + hi * 2;
  v8f c = {};
  for (int k = 0; k < LD; k += 4) {
    v2f a = *(const v2f*)(ap + k);
    int kk = k + hi * 2;
    v2f b;
    b.x = W2[(size_t)kk * DIM + col];
    b.y = W2[(size_t)(kk + 1) * DIM + col];
    c = wmma_f32(a, b, c);
  }
#pragma unroll
  for (int r = 0; r < 8; ++r) {
    int rr = mT * 16 + r + hi * 8;
    qry[(size_t)rr * DIM + col] = c[r] + b2[col];
  }
}

// ---------------------------------------------------------- l2norm --------
__global__ __launch_bounds__(128) void k_l2norm(const float* __restrict__ x,
                                                float* __restrict__ y) {
  __shared__ float red[DIM];
  __shared__ float rs;
  int m = blockIdx.x, t = threadIdx.x;
  float v = x[(size_t)m * DIM + t];
  red[t] = v * v;
  __syncthreads();
  for (int s = 64; s > 0; s >>= 1) {
    if (t < s) red[t] += red[t + s];
    __syncthreads();
  }
  if (t == 0) rs = rsqrtf(red[0] + 1e-12f);
  __syncthreads();
  y[(size_t)m * DIM + t] = v * rs;
}

// --------------------------------------- sims passes (MODE 0/1) -----------
// Block: CHUNK items x all 64 rows, 8 waves. Double-buffered async staging of
// 32-item x 128-float tiles into padded LDS; each wave computes one 16x16
// WMMA tile per stage (nT = wave&1, mT = wave>>1).
// MODE 0: per-row 64-bin histogram.  MODE 1: >= threshold candidate push.
template <int MODE>
__global__ __launch_bounds__(256) void k_sims(
    const float* __restrict__ qn, const float* __restrict__ emb,
    unsigned int* __restrict__ ghist, const float* __restrict__ thresh,
    int* __restrict__ cnt, float* __restrict__ candV, int* __restrict__ candI,
    int nItems) {
  __shared__ float sE[2][TI * EPAD];     // 2 x 16.5KB staged embedding tiles
  __shared__ unsigned int hist[BQ * NBIN];
  __shared__ float sThr[BQ];

  int t = threadIdx.x;
  int lane = t & 31, wave = t >> 5;
  int l15 = lane & 15, hi = lane >> 4;
  int nT = wave & 1, mT = wave >> 1;
  long long blockBase = (long long)blockIdx.x * CHUNK;

  if (MODE == 0) {
    for (int i = t; i < BQ * NBIN; i += 256) hist[i] = 0u;
  } else {
    if (t < BQ) sThr[t] = thresh[t];
  }

  // staging role: thread t copies 64B of row (t>>3), sub-chunk (t&7)
  int sRow = t >> 3, sSub = t & 7;

  const int NSTAGE = CHUNK / TI;  // 32
  {
    long long gr = blockBase + sRow;
    if (gr >= nItems) gr = 0;  // clamp OOB source (data discarded later)
    async_copy64((const char*)(emb + (size_t)gr * DIM) + sSub * 64,
                 (char*)&sE[0][sRow * EPAD] + sSub * 64);
  }

  for (int s = 0; s < NSTAGE; ++s) {
    const float* cur = sE[s & 1];
    float* nxt = sE[(s + 1) & 1];
    if (s + 1 < NSTAGE) {
      long long gr = blockBase + (long long)(s + 1) * TI + sRow;
      if (gr >= nItems) gr = 0;
      async_copy64((const char*)(emb + (size_t)gr * DIM) + sSub * 64,
                   (char*)&nxt[sRow * EPAD] + sSub * 64);
      asm volatile("s_wait_asynccnt 0x4" ::: "memory");  // prev 4 done (in-order)
    } else {
      asm volatile("s_wait_asynccnt 0x0" ::: "memory");
    }
    __syncthreads();  // staged tile visible to all waves

    long long stageBase = blockBase + (long long)s * TI;
    long long item = stageBase + nT * 16 + l15;
    bool valid = item < nItems;
    int row = mT * 16 + l15;
    const float* ap = qn + (size_t)row * DIM + hi * 2;
    const float* bp = cur + (nT * 16 + l15) * EPAD + hi * 2;  // conflict-free
    v8f c = {};
    float nacc = 0.0f;
#pragma unroll 4
    for (int k = 0; k < DIM; k += 4) {
      v2f a = *(const v2f*)(ap + k);
      v2f b = *(const v2f*)(bp + k);   // ds_load_b64
      nacc += b.x * b.x + b.y * b.y;
      c = wmma_f32(a, b, c);           // uniform control flow: EXEC all-ones
    }
    nacc += __shfl_xor(nacc, 16, 32);
    float rn = rsqrtf(nacc + 1e-12f);
#pragma unroll
    for (int r = 0; r < 8; ++r) {
      int rrow = mT * 16 + r + hi * 8;
      float sim = c[r] * rn;
      if (item == 0) sim = -INFINITY;  // sims[:,0] = -inf
      if (MODE == 0) {
        if (valid) {
          int bin = (int)floorf((sim + 1.0f) * 32.0f);
          bin = min(max(bin, 0), NBIN - 1);
          atomicAdd(&hist[rrow * NBIN + bin], 1u);
        }
      } else {
        if (valid && sim >= sThr[rrow]) {
          int pos = atomicAdd(&cnt[rrow], 1);
          if (pos < CAP) {
            candV[(size_t)rrow * CAP + pos] = sim;
            candI[(size_t)rrow * CAP + pos] = (int)item;
          }
        }
      }
    }
    __syncthreads();  // done reading cur before it is overwritten
  }

  if (MODE == 0) {
    for (int i = t; i < BQ * NBIN; i += 256)
      if (hist[i]) atomicAdd(&ghist[i], hist[i]);
  }
}

// ------------------------------------------------ per-row threshold -------
__global__ __launch_bounds__(64) void k_thresh(const unsigned int* __restrict__ hist,
                                               float* __restrict__ thresh) {
  int m = threadIdx.x;
  if (m >= BQ) return;
  int cum = 0;
  float th = -2.0f;  // collect-everything fallback
  for (int b = NBIN - 1; b >= 0; --b) {
    cum += (int)hist[m * NBIN + b];
    if (cum >= MAXK) { th = (float)b * (1.0f / 32.0f) - 1.0f; break; }
  }
  thresh[m] = th;
}

// --------------------------------- final select + fixed_idx + losses ------
__global__ __launch_bounds__(128) void k_finalize(
    const float* __restrict__ emb, const float* __restrict__ linear,
    const int* __restrict__ target, const float* __restrict__ candV,
    const int* __restrict__ candI, const int* __restrict__ cnt,
    int* __restrict__ outIdx, float* __restrict__ outLoss) {
  __shared__ float sv[CAP];
  __shared__ int si[CAP];
  __shared__ float red[128];
  __shared__ int redp[128];
  __shared__ int mkIdx[MAXK];
  __shared__ int fixedIdx[TOPK];
  __shared__ float logits[TOPK];
  __shared__ int sTp;

  int m = blockIdx.x, t = threadIdx.x;
  int c = min(cnt[m], CAP);
  for (int i = t; i < c; i += 128) {
    sv[i] = candV[(size_t)m * CAP + i];
    si[i] = candI[(size_t)m * CAP + i];
  }
  __syncthreads();

  // 50 sequential argmax extractions -> sorted (descending) top-50.
  for (int k = 0; k < MAXK; ++k) {
    float bv = -INFINITY;
    int bp = -1;
    for (int i = t; i < c; i += 128)
      if (sv[i] > bv) { bv = sv[i]; bp = i; }
    red[t] = bv; redp[t] = bp;
    __syncthreads();
    for (int s = 64; s > 0; s >>= 1) {
      if (t < s && red[t + s] > red[t]) { red[t] = red[t + s]; redp[t] = redp[t + s]; }
      __syncthreads();
    }
    if (t == 0) {
      int p = redp[0];
      if (p >= 0) { mkIdx[k] = si[p]; sv[p] = -INFINITY; }
      else         { mkIdx[k] = 0; }
    }
    __syncthreads();
  }

  int tgt = target[m];
  if (t == 0) {
    bool contains = false;
    for (int j = 0; j < TOPK; ++j) {
      fixedIdx[j] = mkIdx[j];
      if (mkIdx[j] == tgt) contains = true;
    }
    if (!contains) fixedIdx[TOPK - 1] = tgt;
    for (int j = 0; j < TOPK; ++j) outIdx[m * TOPK + j] = fixedIdx[j];
    int tp = 0;
    for (int j = TOPK - 1; j >= 0; --j)
      if (fixedIdx[PERM20[j]] == tgt) tp = j;
    sTp = tp;
  }
  __syncthreads();

  // logits[j] = emb[mkIdx[PERM[j]]] . linear   (perm_emb uses pre-fix top_k_emb)
  for (int j = 0; j < TOPK; ++j) {
    long long e = mkIdx[PERM20[j]];
    red[t] = emb[e * DIM + t] * linear[t];
    __syncthreads();
    for (int s = 64; s > 0; s >>= 1) {
      if (t < s) red[t] += red[t + s];
      __syncthreads();
    }
    if (t == 0) logits[j] = red[0];
    __syncthreads();
  }

  // contrastive dots: target_emb . perm_emb[j] for j != target_pos
  float clAcc = 0.0f;
  for (int j = 0; j < TOPK; ++j) {
    if (j == sTp) continue;
    long long e = mkIdx[PERM20[j]];
    red[t] = emb[(long long)tgt * DIM + t] * emb[e * DIM + t];
    __syncthreads();
    for (int s = 64; s > 0; s >>= 1) {
      if (t < s) red[t] += red[t + s];
      __syncthreads();
    }
    if (t == 0) {
      float d = red[0];
      clAcc += 1.0f - 1.0f / (1.0f + __expf(-d));
    }
    __syncthreads();
  }

  if (t == 0) {
    float mx = logits[0];
    for (int j = 1; j < TOPK; ++j) mx = fmaxf(mx, logits[j]);
    float se = 0.0f;
    for (int j = 0; j < TOPK; ++j) se += __expf(logits[j] - mx);
    float ce = -(logits[sTp] - mx - __logf(se));
    float cl = clAcc / (float)(TOPK - 1);
    atomicAdd(outLoss, (ce + cl) * (1.0f / (float)BQ));
  }
}

// ---------------------------------------------------------------------------
extern "C" void kernel_launch(void* const* d_in, const int* in_sizes, int n_in,
                              void* d_out, int out_size, void* d_ws, size_t ws_size,
                              hipStream_t stream) {
  const float* q    = (const float*)d_in[0];
  const float* W1   = (const float*)d_in[1];
  const float* b1   = (const float*)d_in[2];
  const float* W2   = (const float*)d_in[3];
  const float* b2   = (const float*)d_in[4];
  const float* emb  = (const float*)d_in[5];
  const float* lin  = (const float*)d_in[6];
  const int*   tgt  = (const int*)d_in[7];
  int nItems = in_sizes[5] / DIM;

  // workspace carve (256B aligned)
  char* w = (char*)d_ws;
  size_t off = 0;
  auto carve = [&](size_t bytes) {
    void* p = w + off;
    off += (bytes + 255) & ~(size_t)255;
    return p;
  };
  float*        ws_h      = (float*)carve((size_t)BQ * LD * 4);
  float*        ws_query  = (float*)carve((size_t)BQ * DIM * 4);
  float*        ws_qn     = (float*)carve((size_t)BQ * DIM * 4);
  unsigned int* ws_hist   = (unsigned int*)carve((size_t)BQ * NBIN * 4);
  float*        ws_thresh = (float*)carve((size_t)BQ * 4);
  int*          ws_cnt    = (int*)carve((size_t)BQ * 4);
  float*        ws_candV  = (float*)carve((size_t)BQ * CAP * 4);
  int*          ws_candI  = (int*)carve((size_t)BQ * CAP * 4);

  int*   outIdx  = (int*)d_out;                       // 64*20 int32
  float* outLoss = (float*)d_out + BQ * TOPK;         // 1 float

  k_init<<<(BQ * NBIN + 255) / 256, 256, 0, stream>>>(ws_hist, ws_cnt, outLoss);

  k_mlp1<<<dim3(LD / 16, BQ / 16), 32, 0, stream>>>(q, W1, b1, ws_h);
  k_mlp2<<<dim3(DIM / 16, BQ / 16), 32, 0, stream>>>(ws_h, W2, b2, ws_query);
  k_l2norm<<<BQ, DIM, 0, stream>>>(ws_query, ws_qn);

  int nChunks = (nItems + CHUNK - 1) / CHUNK;
  k_sims<0><<<nChunks, 256, 0, stream>>>(ws_qn, emb, ws_hist, nullptr,
                                         nullptr, nullptr, nullptr, nItems);
  k_thresh<<<1, 64, 0, stream>>>(ws_hist, ws_thresh);
  k_sims<1><<<nChunks, 256, 0, stream>>>(ws_qn, emb, nullptr, ws_thresh,
                                         ws_cnt, ws_candV, ws_candI, nItems);
  k_finalize<<<BQ, 128, 0, stream>>>(emb, lin, tgt, ws_candV, ws_candI, ws_cnt,
                                     outIdx, outLoss);
}